// TAGCN_48241072669148
// MI455X (gfx1250) — compile-verified
//
#include <hip/hip_runtime.h>
#include <hip/hip_bf16.h>
#include <math.h>

typedef float v2f __attribute__((ext_vector_type(2)));
typedef float v8f __attribute__((ext_vector_type(8)));

#define D_IN   128
#define D_HID  128
#define D_OUT  40
#define D_OUTP 48   // padded layer-2 width (3 x 16 WMMA tiles)

// ---------------------------------------------------------------- utilities

__device__ __forceinline__ void atomAddF32(float* p, float v) {
  // Lowers to global_atomic_add_f32 (HW f32 atomic, CDNA5)
  unsafeAtomicAdd(p, v);
}

__global__ void zero_f32_kernel(float* __restrict__ p, size_t n) {
  size_t i = (size_t)blockIdx.x * blockDim.x + threadIdx.x;
  size_t stride = (size_t)gridDim.x * blockDim.x;
  for (; i < n; i += stride) p[i] = 0.0f;
}

// deg[dst] += w  (atomic)
__global__ void deg_kernel(const int* __restrict__ dst, const float* __restrict__ w,
                           float* __restrict__ deg, int E) {
  int e = blockIdx.x * blockDim.x + threadIdx.x;
  if (e < E) atomAddF32(deg + dst[e], w[e]);
}

// deg -> dis = (deg>0) ? rsqrt(deg) : 0   (in place)
__global__ void dis_kernel(float* __restrict__ deg, int n) {
  int i = blockIdx.x * blockDim.x + threadIdx.x;
  if (i < n) {
    float d = deg[i];
    deg[i] = (d > 0.0f) ? rsqrtf(d) : 0.0f;
  }
}

// norm[e] = dis[src]*w*dis[dst]
__global__ void norm_kernel(const int* __restrict__ src, const int* __restrict__ dst,
                            const float* __restrict__ w, const float* __restrict__ dis,
                            float* __restrict__ norm, int E) {
  int e = blockIdx.x * blockDim.x + threadIdx.x;
  if (e < E) norm[e] = dis[src[e]] * w[e] * dis[dst[e]];
}

// W2 (4 x 128 x 40) -> W2p (4 x 128 x 48), zero padded
__global__ void pad_w2_kernel(const float* __restrict__ W2, float* __restrict__ W2p) {
  int i = blockIdx.x * blockDim.x + threadIdx.x;   // over 4*128*48
  if (i < 4 * D_HID * D_OUTP) {
    int col = i % D_OUTP;
    int r = i / D_OUTP;                            // [0, 512)
    W2p[i] = (col < D_OUT) ? W2[r * D_OUT + col] : 0.0f;
  }
}

// hout[dst] += norm * hin[src], one wave32 per edge, float4 per lane (128 feats)
__global__ void __launch_bounds__(256)
scatter_kernel(const int* __restrict__ src, const int* __restrict__ dst,
               const float* __restrict__ norm, const float* __restrict__ hin,
               float* __restrict__ hout, int E) {
  int wave = blockIdx.x * (blockDim.x >> 5) + (threadIdx.x >> 5);
  if (wave >= E) return;
  int lane = threadIdx.x & 31;
  int s = src[wave];
  int d = dst[wave];
  float nw = norm[wave];
  const float4* in4 = (const float4*)(hin + (size_t)s * D_IN);
  float4 v = in4[lane];
  float* op = hout + (size_t)d * D_IN + lane * 4;
  atomAddF32(op + 0, v.x * nw);
  atomAddF32(op + 1, v.y * nw);
  atomAddF32(op + 2, v.z * nw);
  atomAddF32(op + 3, v.w * nw);
}

// ---------------------------------------------------------------- WMMA GEMM
// C[M x DoutP] (+)= A[M x 128] * B[128 x DoutP], all row-major, DoutP = NT*16
// compile-time (128 for layer 1, 48 padded for layer 2) so there is ZERO lane
// divergence: EXEC is all-1 throughout, no guards around WMMA.
//
// B (<=64 KB) is staged once per 256-thread block into LDS via the gfx1250
// async-to-LDS path (ASYNCcnt), then WMMA B-fragments come from ds_load.
//
// Fragments for V_WMMA_F32_16X16X4_F32:
//   A (16x4):  lane 0-15 -> M=lane (v0=K0,v1=K1); lane 16-31 -> (K2,K3)
//   B (4x16):  lane -> N (lane&15); K = (lane>>4)*2 + vgpr
//   C/D 16x16: lane -> N; 8 VGPRs -> M = (lane>>4)*8 + i
template <int NT>
__global__ void __launch_bounds__(256)
gemm_tag_kernel(const float* __restrict__ A, const float* __restrict__ B,
                float* __restrict__ C, int M, int accumulate) {
  constexpr int DoutP = NT * 16;
  __shared__ float ldsB[D_IN * DoutP];

  // ---- stage B into LDS (async DMA, 16B per lane-op), whole block cooperates
  {
    const int tid = threadIdx.x;
    const int nVec4 = (D_IN * DoutP) / 4;
#if defined(__HIP_DEVICE_COMPILE__)
    unsigned int ldsBase = (unsigned int)(uintptr_t)(&ldsB[0]);
    for (int i = tid; i < nVec4; i += 256) {
      unsigned long long ga = (unsigned long long)(uintptr_t)(B + (size_t)i * 4);
      unsigned int la = ldsBase + (unsigned int)i * 16u;
      asm volatile("global_load_async_to_lds_b128 %0, %1, off"
                   :: "v"(la), "v"(ga) : "memory");
    }
    asm volatile("s_wait_asynccnt 0" ::: "memory");
#else
    for (int i = tid; i < nVec4; i += 256)
      ((float4*)ldsB)[i] = ((const float4*)B)[i];
#endif
  }
  __syncthreads();

  const int lane = threadIdx.x & 31;
  const int wave = blockIdx.x * (blockDim.x >> 5) + (threadIdx.x >> 5);
  const int M0 = wave << 4;
  if (M0 >= M) return;                       // wave-uniform
  const int halfSel = (lane >> 4) << 1;      // 0 or 2
  const int mRow = M0 + (lane & 15);
  const int nLane = lane & 15;
  const int rBase = M0 + ((lane >> 4) << 3); // C rows for this lane-half

  v8f acc[NT];
#pragma unroll
  for (int j = 0; j < NT; ++j) {
    const int col = j * 16 + nLane;
    if (accumulate) {
#pragma unroll
      for (int i = 0; i < 8; ++i)
        acc[j][i] = C[(size_t)(rBase + i) * DoutP + col];
    } else {
#pragma unroll
      for (int i = 0; i < 8; ++i) acc[j][i] = 0.0f;
    }
  }

  const float* aRow = A + (size_t)mRow * D_IN;
#pragma unroll 4
  for (int kk = 0; kk < D_IN; kk += 4) {
    const float2 av = *(const float2*)(aRow + kk + halfSel);
    v2f a;
    a.x = av.x;
    a.y = av.y;
    const float* bk = &ldsB[(kk + halfSel) * DoutP + nLane];
#pragma unroll
    for (int j = 0; j < NT; ++j) {
      v2f b;
      b.x = bk[j * 16];
      b.y = bk[j * 16 + DoutP];
      acc[j] = __builtin_amdgcn_wmma_f32_16x16x4_f32(
          false, a, false, b, (short)0, acc[j], false, false);
    }
  }

#pragma unroll
  for (int j = 0; j < NT; ++j) {
    const int col = j * 16 + nLane;
#pragma unroll
    for (int i = 0; i < 8; ++i)
      C[(size_t)(rBase + i) * DoutP + col] = acc[j][i];
  }
}

// out = relu(out + b[col])   in place over N x 128, float4 vectorized
__global__ void bias_relu_kernel(float* __restrict__ x, const float* __restrict__ b,
                                 size_t n4) {
  size_t i = (size_t)blockIdx.x * blockDim.x + threadIdx.x;
  if (i < n4) {
    float4 v = ((float4*)x)[i];
    const float4 bb = ((const float4*)b)[i % (D_HID / 4)];
    v.x = fmaxf(v.x + bb.x, 0.0f);
    v.y = fmaxf(v.y + bb.y, 0.0f);
    v.z = fmaxf(v.z + bb.z, 0.0f);
    v.w = fmaxf(v.w + bb.w, 0.0f);
    ((float4*)x)[i] = v;
  }
}

// out[N x 40] = log_softmax(in[N x 48 padded] + b), one wave per row
__global__ void __launch_bounds__(256)
bias_logsoftmax_kernel(const float* __restrict__ in, const float* __restrict__ b,
                       float* __restrict__ out, int nRows) {
  int row = blockIdx.x * (blockDim.x >> 5) + (threadIdx.x >> 5);
  if (row >= nRows) return;
  int lane = threadIdx.x & 31;
  const float* rp = in + (size_t)row * D_OUTP;
  float* op = out + (size_t)row * D_OUT;
  const bool has2 = (lane + 32) < D_OUT;
  float v0 = rp[lane] + b[lane];
  float v1 = has2 ? (rp[lane + 32] + b[lane + 32]) : -INFINITY;

  float m = fmaxf(v0, v1);
#pragma unroll
  for (int off = 16; off >= 1; off >>= 1) m = fmaxf(m, __shfl_xor(m, off, 32));

  float s = __expf(v0 - m) + (has2 ? __expf(v1 - m) : 0.0f);
#pragma unroll
  for (int off = 16; off >= 1; off >>= 1) s += __shfl_xor(s, off, 32);

  float lse = m + __logf(s);
  op[lane] = v0 - lse;
  if (has2) op[lane + 32] = v1 - lse;
}

// ---------------------------------------------------------------- launcher

extern "C" void kernel_launch(void* const* d_in, const int* in_sizes, int n_in,
                              void* d_out, int out_size, void* d_ws, size_t ws_size,
                              hipStream_t stream) {
  const float* x  = (const float*)d_in[0];        // N x 128
  const int*  ei  = (const int*)d_in[1];          // 2 x E
  const float* ew = (const float*)d_in[2];        // E
  const float* W1 = (const float*)d_in[3];        // 4 x 128 x 128
  const float* b1 = (const float*)d_in[4];        // 128
  const float* W2 = (const float*)d_in[5];        // 4 x 128 x 40
  const float* b2 = (const float*)d_in[6];        // 40
  float* out = (float*)d_out;                     // N x 40

  const int N = in_sizes[0] / D_IN;
  const int E = in_sizes[2];
  const int* src = ei;
  const int* dst = ei + E;

  // workspace carving (64-element aligned)
  float* deg   = (float*)d_ws;                             // N
  float* norm  = deg + (((size_t)N + 63) & ~(size_t)63);   // E
  float* bufA  = norm + (((size_t)E + 63) & ~(size_t)63);  // N*128
  float* bufB  = bufA + (size_t)N * D_IN;                  // N*128
  float* out1  = bufB + (size_t)N * D_IN;                  // N*128
  float* out2p = out1 + (size_t)N * D_IN;                  // N*48
  float* W2p   = out2p + (size_t)N * D_OUTP;               // 4*128*48
  (void)ws_size;

  const size_t nFeat = (size_t)N * D_IN;
  const int TB = 256;
  const int blkN     = (N + TB - 1) / TB;
  const int blkE     = (E + TB - 1) / TB;
  const int blkFeat  = (int)((nFeat + TB - 1) / TB);
  const int blkFeat4 = (int)((nFeat / 4 + TB - 1) / TB);
  const int blkEdgeW = (E + 7) / 8;           // 8 waves/block, wave per edge
  const int nTilesM  = (N + 15) / 16;
  const int blkGemm  = (nTilesM + 7) / 8;     // 8 waves/block, wave per M-tile
  const int blkRowW  = (N + 7) / 8;           // 8 waves/block, wave per row
  const int blkPad   = (4 * D_HID * D_OUTP + TB - 1) / TB;

  // ---- normalization
  zero_f32_kernel<<<blkN, TB, 0, stream>>>(deg, (size_t)N);
  deg_kernel<<<blkE, TB, 0, stream>>>(dst, ew, deg, E);
  dis_kernel<<<blkN, TB, 0, stream>>>(deg, N);
  norm_kernel<<<blkE, TB, 0, stream>>>(src, dst, ew, deg, norm, E);
  pad_w2_kernel<<<blkPad, TB, 0, stream>>>(W2, W2p);

  // ---- layer 1: out1 = sum_k A^k x W1[k]
  gemm_tag_kernel<8><<<blkGemm, TB, 0, stream>>>(x, W1, out1, N, 0);

  zero_f32_kernel<<<blkFeat, TB, 0, stream>>>(bufA, nFeat);
  scatter_kernel<<<blkEdgeW, TB, 0, stream>>>(src, dst, norm, x, bufA, E);
  gemm_tag_kernel<8><<<blkGemm, TB, 0, stream>>>(bufA, W1 + 1 * D_IN * D_HID, out1, N, 1);

  zero_f32_kernel<<<blkFeat, TB, 0, stream>>>(bufB, nFeat);
  scatter_kernel<<<blkEdgeW, TB, 0, stream>>>(src, dst, norm, bufA, bufB, E);
  gemm_tag_kernel<8><<<blkGemm, TB, 0, stream>>>(bufB, W1 + 2 * D_IN * D_HID, out1, N, 1);

  zero_f32_kernel<<<blkFeat, TB, 0, stream>>>(bufA, nFeat);
  scatter_kernel<<<blkEdgeW, TB, 0, stream>>>(src, dst, norm, bufB, bufA, E);
  gemm_tag_kernel<8><<<blkGemm, TB, 0, stream>>>(bufA, W1 + 3 * D_IN * D_HID, out1, N, 1);

  // h = relu(out1 + b1), in place
  bias_relu_kernel<<<blkFeat4, TB, 0, stream>>>(out1, b1, nFeat / 4);

  // ---- layer 2: out2p = sum_k A^k h W2p[k]   (padded to 48 cols)
  gemm_tag_kernel<3><<<blkGemm, TB, 0, stream>>>(out1, W2p, out2p, N, 0);

  zero_f32_kernel<<<blkFeat, TB, 0, stream>>>(bufB, nFeat);
  scatter_kernel<<<blkEdgeW, TB, 0, stream>>>(src, dst, norm, out1, bufB, E);
  gemm_tag_kernel<3><<<blkGemm, TB, 0, stream>>>(bufB, W2p + 1 * D_HID * D_OUTP, out2p, N, 1);

  zero_f32_kernel<<<blkFeat, TB, 0, stream>>>(bufA, nFeat);
  scatter_kernel<<<blkEdgeW, TB, 0, stream>>>(src, dst, norm, bufB, bufA, E);
  gemm_tag_kernel<3><<<blkGemm, TB, 0, stream>>>(bufA, W2p + 2 * D_HID * D_OUTP, out2p, N, 1);

  zero_f32_kernel<<<blkFeat, TB, 0, stream>>>(bufB, nFeat);
  scatter_kernel<<<blkEdgeW, TB, 0, stream>>>(src, dst, norm, bufA, bufB, E);
  gemm_tag_kernel<3><<<blkGemm, TB, 0, stream>>>(bufB, W2p + 3 * D_HID * D_OUTP, out2p, N, 1);

  // ---- epilogue: out = log_softmax(out2p + b2)
  bias_logsoftmax_kernel<<<blkRowW, TB, 0, stream>>>(out2p, b2, out, N);
}